// BiLSTM_CRF_49048526520834
// MI455X (gfx1250) — compile-verified
//
#include <hip/hip_runtime.h>
#include <math.h>
#include <stdint.h>

// ---------------------------------------------------------------------------
// BiLSTM-CRF for MI455X (gfx1250), wave32, f32 WMMA (V_WMMA_F32_16X16X4_F32).
//
// Pipeline:
//   1) k_gemm_z   : Z[t*B+b, :512] = emb[sent[b,t]] @ W_ih^T + bias   (x2 dirs)
//                   B-tile staged into LDS by the Tensor Data Mover (TDM)
//                   with hardware bank-padding; A-tile gathered via b128 loads.
//   2) k_lstm     : sequential recurrence, batch split across 8 WGs,
//                   h/c resident in LDS, W_hh from L2/WGP$, prefetch of Z.
//   3) k_emis_crf : emission = h_cat @ W_out^T + b_out, expanded with
//                   transition[i,j] into crf_scores[b,t,i,j].
// ---------------------------------------------------------------------------

typedef __attribute__((ext_vector_type(2))) float v2f;
typedef __attribute__((ext_vector_type(8))) float v8f;

#if __has_builtin(__builtin_amdgcn_tensor_load_to_lds)
#define HAVE_TDM 1
typedef __attribute__((ext_vector_type(4))) unsigned int u32x4;
typedef __attribute__((ext_vector_type(4))) int          i32x4;
typedef __attribute__((ext_vector_type(8))) int          i32x8;
#else
#define HAVE_TDM 0
#endif

#define TSEQ  512
#define BATCH 64
#define EMBD  256
#define HDIM  128             // per-direction hidden
#define G4    512             // 4*HDIM
#define HID2  256             // bidirectional hidden
#define NTAGS 16
#define MROWS (TSEQ * BATCH)  // 32768

static __device__ __forceinline__ v8f wmma_f32(v2f a, v2f b, v8f c) {
  // 8 args: (neg_a, A, neg_b, B, c_mod, C, reuse_a, reuse_b)
  return __builtin_amdgcn_wmma_f32_16x16x4_f32(false, a, false, b,
                                               (short)0, c, false, false);
}

static __device__ __forceinline__ float sigm(float x) {
  return 1.0f / (1.0f + expf(-x));
}

// ---------------------------------------------------------------------------
// Kernel 1: gather embeddings + input GEMM (M=32768, K=256, N=512)
// WG tile: 128(M) x 64(N), 8 waves, each wave 16(M) x 64(N) (4 accumulators).
// K chunked by 64 through LDS. B tile moved by TDM (tensor_load_to_lds) with
// pad_interval=64 dwords / pad_amount=4 dwords -> LDS row stride 68 dwords.
// ---------------------------------------------------------------------------
__global__ __launch_bounds__(256) void k_gemm_z(
    const int* __restrict__ sent, const float* __restrict__ emb,
    const float* __restrict__ Wih, const float* __restrict__ bias,
    float* __restrict__ Z)
{
  __shared__ float As[128][68];   // gathered embedding rows, padded stride
  __shared__ float Bs[64][68];    // W_ih rows (B matrix is W^T, K x N)

  const int tid  = threadIdx.x;
  const int lane = tid & 31;
  const int w    = tid >> 5;      // wave 0..7
  const int ln   = lane & 15;
  const int hi   = lane >> 4;     // selects K pair (k, k+1) vs (k+2, k+3)
  const int mBlock = blockIdx.x * 128;
  const int nBase  = blockIdx.y * 64;

  v8f acc[4];
#pragma unroll
  for (int j = 0; j < 4; ++j)
#pragma unroll
    for (int e = 0; e < 8; ++e) acc[j][e] = 0.0f;

  // A-tile load task: 256 threads = 128 rows x 2 halves (32 floats each)
  const int arow  = tid >> 1;
  const int ahalf = tid & 1;
  const int am    = mBlock + arow;            // m = t*B + b
  const int tok   = sent[(am & (BATCH - 1)) * TSEQ + (am >> 6)];
  const float* aSrc = emb + (size_t)tok * EMBD + ahalf * 32;

#if !HAVE_TDM
  // Fallback B-tile load task: 64 rows x 4 quarters (16 floats each)
  const int brow = tid >> 2;
  const int bq   = tid & 3;
  const float* bSrc = Wih + (size_t)(nBase + brow) * EMBD + bq * 16;
#endif

  for (int kc = 0; kc < EMBD; kc += 64) {
    __syncthreads();

#if HAVE_TDM
    if (w == 0) {
      // ---- Tensor DMA descriptor (D#), per CDNA5 ISA ----
      const unsigned long long ga =
          (unsigned long long)(uintptr_t)(Wih + (size_t)nBase * EMBD + kc);
      const unsigned ldsAddr = (unsigned)(uintptr_t)&Bs[0][0];
      u32x4 g0;
      g0[0] = 1u;                                   // count=1, no gather
      g0[1] = ldsAddr;                              // lds_addr (bytes)
      g0[2] = (unsigned)ga;                         // global_addr[31:0]
      g0[3] = (unsigned)(ga >> 32) | (2u << 30);    // global_addr[56:32] | type=2
      i32x8 g1;
      g1[0] = (2 << 16)       // data_size = 4 bytes
            | (1 << 20)       // pad_enable
            | (5 << 22)       // pad_interval: 64 dwords
            | (3 << 25);      // pad_amount: 4 dwords -> LDS stride 68
      g1[1] = (EMBD & 0xFFFF) << 16;  // tensor_dim0[15:0] at bits[63:48]
      g1[2] = (G4 & 0xFFFF) << 16;    // tensor_dim0 hi=0 | tensor_dim1[15:0]
      g1[3] = (64 << 16);             // tensor_dim1 hi=0 | tile_dim0 = 64
      g1[4] = 64;                     // tile_dim1 = 64 rows | tile_dim2 = 0
      g1[5] = EMBD;                   // tensor_dim0_stride[31:0] = 256
      g1[6] = 0;                      // stride hi | tensor_dim1_stride lo
      g1[7] = 0;
      i32x4 gz;
      gz[0] = 0; gz[1] = 0; gz[2] = 0; gz[3] = 0;
#if __clang_major__ >= 23
      i32x8 gz8;
#pragma unroll
      for (int e = 0; e < 8; ++e) gz8[e] = 0;
      __builtin_amdgcn_tensor_load_to_lds(g0, g1, gz, gz, gz8, 0);
#else
      __builtin_amdgcn_tensor_load_to_lds(g0, g1, gz, gz, 0);
#endif
      __builtin_amdgcn_s_wait_tensorcnt(0);
    }
#else
#pragma unroll
    for (int u = 0; u < 4; ++u)
      *(float4*)&Bs[brow][bq * 16 + u * 4] = *(const float4*)(bSrc + kc + u * 4);
#endif

#pragma unroll
    for (int u = 0; u < 8; ++u)
      *(float4*)&As[arow][ahalf * 32 + u * 4] = *(const float4*)(aSrc + kc + u * 4);
    __syncthreads();

#pragma unroll
    for (int kk = 0; kk < 64; kk += 4) {
      v2f a = *(const v2f*)&As[w * 16 + ln][kk + 2 * hi];
#pragma unroll
      for (int j = 0; j < 4; ++j) {
        v2f b = *(const v2f*)&Bs[j * 16 + ln][kk + 2 * hi];
        acc[j] = wmma_f32(a, b, acc[j]);
      }
    }
  }

  const int mWave = mBlock + w * 16;
#pragma unroll
  for (int j = 0; j < 4; ++j) {
    const int n  = nBase + j * 16 + ln;
    const float bn = bias[n];
#pragma unroll
    for (int v = 0; v < 8; ++v) {
      const int r = (lane < 16) ? v : v + 8;   // D row layout
      Z[(size_t)(mWave + r) * G4 + n] = acc[j][v] + bn;
    }
  }
}

// ---------------------------------------------------------------------------
// Kernel 2: LSTM recurrence. grid = 8 WGs: (dir 0/1) x (batch quarter 0..3).
// Each WG owns 16 batch rows; per step GEMM [16,128]@[128,512] with 16 waves
// (2 N-tiles each), gates elementwise from LDS, h/c stay in LDS.
// ---------------------------------------------------------------------------
__global__ __launch_bounds__(512) void k_lstm(
    const float* __restrict__ zf, const float* __restrict__ zb,
    const float* __restrict__ WhhF, const float* __restrict__ WhhB,
    float* __restrict__ hcat)
{
  __shared__ float hBuf[16][132];
  __shared__ float cBuf[16][132];
  __shared__ float zBuf[16][516];

  const int tid  = threadIdx.x;
  const int lane = tid & 31;
  const int w    = tid >> 5;           // wave 0..15
  const int ln   = lane & 15;
  const int hi   = lane >> 4;
  const int dir   = (int)(blockIdx.x >> 2);
  const int bBase = (int)(blockIdx.x & 3) * 16;
  const float* Zpre = dir ? zb : zf;
  const float* Whh  = dir ? WhhB : WhhF;
  const int dOff = dir * HDIM;

  {
    float* hp = &hBuf[0][0];
    float* cp = &cBuf[0][0];
    for (int e = tid; e < 16 * 132; e += 512) { hp[e] = 0.0f; cp[e] = 0.0f; }
  }
  __syncthreads();

  for (int step = 0; step < TSEQ; ++step) {
    const int t = dir ? (TSEQ - 1 - step) : step;
    const float* Zp = Zpre + ((size_t)t * BATCH + bBase) * G4;

    // Prefetch next step's pre-activation slab (hides HBM latency in the
    // serial recurrence). Lowers to global_prefetch_b8.
    if (step + 1 < TSEQ) {
      const int tn = dir ? (TSEQ - 2 - step) : step + 1;
      const float* Zn = Zpre + ((size_t)tn * BATCH + bBase) * G4 + (size_t)tid * 16;
      __builtin_prefetch(Zn, 0, 0);
    }

    // ---- z = h @ Whh^T + Zpre ----
#pragma unroll
    for (int s = 0; s < 2; ++s) {
      const int nb = (w * 2 + s) * 16;
      v8f acc;
#pragma unroll
      for (int e = 0; e < 8; ++e) acc[e] = 0.0f;
      const float* wrow = Whh + (size_t)(nb + ln) * HDIM;   // B[k][n] = Whh[n][k]
#pragma unroll
      for (int k = 0; k < HDIM; k += 4) {
        v2f a = *(const v2f*)&hBuf[ln][k + 2 * hi];
        v2f b = *(const v2f*)(wrow + k + 2 * hi);
        acc = wmma_f32(a, b, acc);
      }
#pragma unroll
      for (int v = 0; v < 8; ++v) {
        const int r = (lane < 16) ? v : v + 8;
        zBuf[r][nb + ln] = acc[v] + Zp[(size_t)r * G4 + nb + ln];
      }
    }
    __syncthreads();

    // ---- gates: 16 rows x 128 cols = 2048 elems, 4 per thread ----
#pragma unroll
    for (int jj = 0; jj < 4; ++jj) {
      const int e = tid + jj * 512;
      const int r = e >> 7;
      const int j = e & 127;
      const float iv = zBuf[r][j];
      const float fv = zBuf[r][j + HDIM];
      const float gv = zBuf[r][j + 2 * HDIM];
      const float ov = zBuf[r][j + 3 * HDIM];
      const float c  = sigm(fv) * cBuf[r][j] + sigm(iv) * tanhf(gv);
      const float h  = sigm(ov) * tanhf(c);
      cBuf[r][j] = c;
      hBuf[r][j] = h;
      hcat[((size_t)t * BATCH + bBase + r) * HID2 + dOff + j] = h;
    }
    __syncthreads();
  }
}

// ---------------------------------------------------------------------------
// Kernel 3: emission GEMM (M=32768, K=256, N=16) + CRF expansion.
// One wave per 16-row M tile; N=16 is exactly one WMMA tile.
// ---------------------------------------------------------------------------
__global__ __launch_bounds__(256) void k_emis_crf(
    const float* __restrict__ hcat, const float* __restrict__ Wout,
    const float* __restrict__ bout, const float* __restrict__ trans,
    float* __restrict__ out)
{
  const int tid  = threadIdx.x;
  const int lane = tid & 31;
  const int w    = tid >> 5;
  const int ln   = lane & 15;
  const int hi   = lane >> 4;
  const int mBase = ((int)blockIdx.x * 8 + w) * 16;

  v8f acc;
#pragma unroll
  for (int e = 0; e < 8; ++e) acc[e] = 0.0f;

  const float* arow = hcat + (size_t)(mBase + ln) * HID2;
  const float* brow = Wout + (size_t)ln * HID2;     // B[k][n] = Wout[n][k]
#pragma unroll
  for (int k = 0; k < HID2; k += 4) {
    v2f a = *(const v2f*)(arow + k + 2 * hi);
    v2f b = *(const v2f*)(brow + k + 2 * hi);
    acc = wmma_f32(a, b, acc);
  }

  float tr[16];
#pragma unroll
  for (int i = 0; i < 16; ++i) tr[i] = trans[i * NTAGS + ln];
  const float bn = bout[ln];

#pragma unroll
  for (int v = 0; v < 8; ++v) {
    const int m = mBase + v + ((lane < 16) ? 0 : 8);
    const int t = m >> 6;             // m = t*B + b
    const int b = m & (BATCH - 1);
    const float e = acc[v] + bn;
    float* op = out + (((size_t)b * TSEQ + t) * NTAGS) * NTAGS + ln;
#pragma unroll
    for (int i = 0; i < 16; ++i) op[i * NTAGS] = e + tr[i];
  }
}

// ---------------------------------------------------------------------------
extern "C" void kernel_launch(void* const* d_in, const int* in_sizes, int n_in,
                              void* d_out, int out_size, void* d_ws, size_t ws_size,
                              hipStream_t stream) {
  (void)in_sizes; (void)n_in; (void)out_size; (void)ws_size;

  const int*   sent  = (const int*)  d_in[0];
  const float* emb   = (const float*)d_in[1];
  const float* WihF  = (const float*)d_in[2];
  const float* WhhF  = (const float*)d_in[3];
  const float* bF    = (const float*)d_in[4];
  const float* WihB  = (const float*)d_in[5];
  const float* WhhB  = (const float*)d_in[6];
  const float* bB    = (const float*)d_in[7];
  const float* Wout  = (const float*)d_in[8];
  const float* bout  = (const float*)d_in[9];
  const float* trans = (const float*)d_in[10];

  // Workspace: Zf (64MB) | Zb (64MB) | h_cat (32MB)
  float* zf   = (float*)d_ws;
  float* zb   = zf + (size_t)MROWS * G4;
  float* hcat = zb + (size_t)MROWS * G4;

  k_gemm_z<<<dim3(MROWS / 128, G4 / 64), 256, 0, stream>>>(sent, emb, WihF, bF, zf);
  k_gemm_z<<<dim3(MROWS / 128, G4 / 64), 256, 0, stream>>>(sent, emb, WihB, bB, zb);
  k_lstm<<<8, 512, 0, stream>>>(zf, zb, WhhF, WhhB, hcat);
  k_emis_crf<<<MROWS / (16 * 8), 256, 0, stream>>>(hcat, Wout, bout, trans, (float*)d_out);
}